// MaxPool2d_G_12506944766364
// MI455X (gfx1250) — compile-verified
//
#include <hip/hip_runtime.h>

// MI455X / gfx1250 single-pass kernel:
//   16x16 box-window-sum (pad 8) expressed as two banded fp32 GEMMs on
//   V_WMMA_F32_16X16X4_F32, wave32 argmax reduction, crop gathered from LDS.
// One wave (32 lanes) per (b,c) slice; 2048 slices total.
// Band-matrix fragments are hoisted out of the tile loops (loop-invariant)
// and built with a single unsigned range-check each, so the steady-state
// inner loop is ~2 ds_loads + 1 v_wmma.

typedef __attribute__((ext_vector_type(2))) float v2f;
typedef __attribute__((ext_vector_type(8))) float v8f;

#define LS 68  // LDS row stride in floats: 16B-aligned float4 rows, conflict-free columns

__global__ __launch_bounds__(32)
void boxsum_argmax_crop_kernel(const float* __restrict__ x, float* __restrict__ out) {
  __shared__ float sx[64 * LS];   // input tile
  __shared__ float s1[64 * LS];   // horizontal window sums

  const int lane = threadIdx.x;   // 0..31 (wave32)
  const int m    = lane & 15;     // row (A) / col (B,C) index within a 16-wide frag
  const int hi   = lane >> 4;     // lane half
  const int kh   = hi * 2;        // K sub-offset inside a K=4 fragment
  const long slice = blockIdx.x;  // b*C + c

  const float* xs = x + slice * 4096;  // 64*64 fp32

  // ---- stage 64x64 tile into LDS (coalesced float4) ----
  const float4* xv = (const float4*)xs;
  for (int it = 0; it < 32; ++it) {
    int chunk = it * 32 + lane;        // 1024 float4 chunks
    int el    = chunk << 2;
    int row   = el >> 6;
    int col   = el & 63;
    *(float4*)&sx[row * LS + col] = xv[chunk];
  }
  __syncthreads();

  // ---- Phase A: S1 = X * B,  B[col][j] = 1 iff j-8 <= col <= j+7 (clamped window) ----
  // B-fragment depends only on (tj, k): hoist out of the tr loop.
  for (int tj = 0; tj < 4; ++tj) {
    const int j = tj * 16 + m;         // output column this lane supplies in B
    v2f bf[16];
    #pragma unroll
    for (int kk = 0; kk < 16; ++kk) {
      const int ka = kk * 4 + kh;
      bf[kk].x = ((unsigned)(ka     - j + 8) < 16u) ? 1.0f : 0.0f;
      bf[kk].y = ((unsigned)(ka + 1 - j + 8) < 16u) ? 1.0f : 0.0f;
    }
    for (int tr = 0; tr < 4; ++tr) {
      const float* arow = &sx[(tr * 16 + m) * LS];
      v8f acc = {};
      #pragma unroll
      for (int kk = 0; kk < 16; ++kk) {
        const int ka = kk * 4 + kh;
        v2f a;
        a.x = arow[ka];
        a.y = arow[ka + 1];
        acc = __builtin_amdgcn_wmma_f32_16x16x4_f32(false, a, false, bf[kk],
                                                    (short)0, acc, false, false);
      }
      #pragma unroll
      for (int p = 0; p < 8; ++p)
        s1[(tr * 16 + p + hi * 8) * LS + tj * 16 + m] = acc[p];
    }
  }
  __syncthreads();

  // ---- Phase B: XSUM = M * S1,  M[i][r] = 1 iff i-8 <= r <= i+7; fused argmax ----
  // A-fragment depends only on (ti, k): hoist out of the tj loop.
  float bestv = -__builtin_inff();
  int   besti = 0x7fffffff;
  for (int ti = 0; ti < 4; ++ti) {
    const int i = ti * 16 + m;         // output row this lane supplies in A
    v2f af[16];
    #pragma unroll
    for (int kk = 0; kk < 16; ++kk) {
      const int ka = kk * 4 + kh;
      af[kk].x = ((unsigned)(ka     - i + 8) < 16u) ? 1.0f : 0.0f;
      af[kk].y = ((unsigned)(ka + 1 - i + 8) < 16u) ? 1.0f : 0.0f;
    }
    for (int tj = 0; tj < 4; ++tj) {
      const float* bcol = &s1[tj * 16 + m];
      v8f acc = {};
      #pragma unroll
      for (int kk = 0; kk < 16; ++kk) {
        const int ka = kk * 4 + kh;
        v2f b;
        b.x = bcol[ka * LS];
        b.y = bcol[(ka + 1) * LS];
        acc = __builtin_amdgcn_wmma_f32_16x16x4_f32(false, af[kk], false, b,
                                                    (short)0, acc, false, false);
      }
      #pragma unroll
      for (int p = 0; p < 8; ++p) {
        const int gi  = ti * 16 + p + hi * 8;
        const int gj  = tj * 16 + m;
        const int idx = (gi << 6) | gj;       // flat row-major index, matches reference
        const float v = acc[p];
        if (v > bestv || (v == bestv && idx < besti)) { bestv = v; besti = idx; }
      }
    }
  }

  // ---- wave32 argmax reduction (first-occurrence tie-break) ----
  #pragma unroll
  for (int off = 16; off > 0; off >>= 1) {
    const float ov = __shfl_xor(bestv, off, 32);
    const int   oi = __shfl_xor(besti, off, 32);
    if (ov > bestv || (ov == bestv && oi < besti)) { bestv = ov; besti = oi; }
  }

  // ---- gather the 16x16 crop around (r,c) straight from the LDS tile ----
  const int r = besti >> 6;
  const int c = besti & 63;
  float* op = out + slice * 256;
  const int e0 = lane * 8;             // 8 consecutive outputs per lane
  float vals[8];
  #pragma unroll
  for (int t = 0; t < 8; ++t) {
    const int e  = e0 + t;
    const int di = e >> 4;
    const int dj = e & 15;
    const int sr = r + di - 8;
    const int sc = c + dj - 8;
    vals[t] = (sr >= 0 && sr < 64 && sc >= 0 && sc < 64) ? sx[sr * LS + sc] : 0.0f;
  }
  *(float4*)&op[e0]     = make_float4(vals[0], vals[1], vals[2], vals[3]);
  *(float4*)&op[e0 + 4] = make_float4(vals[4], vals[5], vals[6], vals[7]);
}

extern "C" void kernel_launch(void* const* d_in, const int* in_sizes, int n_in,
                              void* d_out, int out_size, void* d_ws, size_t ws_size,
                              hipStream_t stream) {
  (void)in_sizes; (void)n_in; (void)d_ws; (void)ws_size; (void)out_size;
  const float* x  = (const float*)d_in[0];   // (16,128,64,64) fp32
  float* out      = (float*)d_out;           // (16,128,16,16) fp32
  const int slices = 16 * 128;               // one wave32 workgroup per (b,c)
  boxsum_argmax_crop_kernel<<<dim3(slices), dim3(32), 0, stream>>>(x, out);
}